// Dihedral_40879498729242
// MI455X (gfx1250) — compile-verified
//
#include <hip/hip_runtime.h>
#include <math.h>

#define N_DIHEDRALS 4000000
#define N_BATCH     256
#define N_TYPES     25
#define TBL         (N_TYPES * N_TYPES * N_TYPES * N_TYPES)  // 390625
#define BLOCK       256
#define ITER        8
#define CHUNK       (BLOCK * ITER)
#define TAB_STRIDE  8                                        // floats per packed entry
#define TAB_BYTES   ((size_t)TBL * TAB_STRIDE * sizeof(float))  // 12.5 MB

struct __attribute__((packed, aligned(4))) F3 { float x, y, z; };

__global__ void dihedral_zero_out(float* __restrict__ out) {
    out[threadIdx.x] = 0.0f;
}

// Precompute per-tuple table: [ksum, k1*cos(th1), k1*sin(th1), k2*cos, k2*sin, k3*cos, k3*sin, pad]
__global__ __launch_bounds__(256)
void dihedral_build_table(const float* __restrict__ thetas,
                          const float* __restrict__ ks,
                          float* __restrict__ tab)
{
    const int idx = blockIdx.x * blockDim.x + threadIdx.x;
    if (idx >= TBL) return;

    const float th0 = thetas[idx];
    const float th1 = thetas[TBL + idx];
    const float th2 = thetas[2 * TBL + idx];
    const float k0  = ks[idx];
    const float k1  = ks[TBL + idx];
    const float k2  = ks[2 * TBL + idx];

    float s0, c0, s1, c1, s2, c2;
    sincosf(th0, &s0, &c0);   // accurate path OK here: only 390K elements
    sincosf(th1, &s1, &c1);
    sincosf(th2, &s2, &c2);

    float4* p = (float4*)(tab + (size_t)idx * TAB_STRIDE);
    p[0] = make_float4(k0 + k1 + k2, k0 * c0, k0 * s0, k1 * c1);
    p[1] = make_float4(k1 * s1, k2 * c2, k2 * s2, 0.0f);
}

// Geometry: cos/sin of the dihedral angle via atan2-free math.
__device__ __forceinline__ void dihedral_cs(const float* __restrict__ pos,
                                            int a0, int a1, int a2, int a3,
                                            float& c1, float& s1)
{
    const F3 p0 = ((const F3*)pos)[a0];   // global_load_b96 gathers
    const F3 p1 = ((const F3*)pos)[a1];
    const F3 p2 = ((const F3*)pos)[a2];
    const F3 p3 = ((const F3*)pos)[a3];

    const float d1x = p1.x - p0.x, d1y = p1.y - p0.y, d1z = p1.z - p0.z;
    const float d2x = p2.x - p1.x, d2y = p2.y - p1.y, d2z = p2.z - p1.z;
    const float d3x = p3.x - p2.x, d3y = p3.y - p2.y, d3z = p3.z - p2.z;

    const float n1x = d1y * d2z - d1z * d2y;
    const float n1y = d1z * d2x - d1x * d2z;
    const float n1z = d1x * d2y - d1y * d2x;
    const float n2x = d2y * d3z - d2z * d3y;
    const float n2y = d2z * d3x - d2x * d3z;
    const float n2z = d2x * d3y - d2y * d3x;

    const float m1ux = n1y * d2z - n1z * d2y;
    const float m1uy = n1z * d2x - n1x * d2z;
    const float m1uz = n1x * d2y - n1y * d2x;

    const float len2  = d2x * d2x + d2y * d2y + d2z * d2z;
    const float invl2 = rsqrtf(fmaxf(len2, 1e-30f));

    const float x = n1x * n2x + n1y * n2y + n1z * n2z;
    const float y = (m1ux * n2x + m1uy * n2y + m1uz * n2z) * invl2;

    const float r2   = x * x + y * y;
    const float invr = rsqrtf(fmaxf(r2, 1e-30f));
    const bool  ok   = r2 > 1e-30f;
    c1 = ok ? x * invr : 1.0f;   // atan2(0,0) == 0 in the reference
    s1 = ok ? y * invr : 0.0f;
}

// Wave-level segmented accumulate into LDS bins. mapping_batch is sorted, so
// waves are batch-uniform except at ~256 boundaries: reduce across the wave
// with shuffles and issue ONE ds_add_f32 instead of 32 serialized ones.
// Requires full EXEC (callers mask V to 0 for out-of-range work items).
__device__ __forceinline__ void wave_bin_add(float* __restrict__ bins, int b, float V)
{
    const int b0 = __shfl(b, 0, 32);
    if (__all(b == b0)) {
        #pragma unroll
        for (int off = 16; off > 0; off >>= 1)
            V += __shfl_xor(V, off, 32);
        if ((threadIdx.x & 31) == 0)
            atomicAdd(&bins[b0], V);          // 1 ds_add_f32 per wave
    } else {
        atomicAdd(&bins[b], V);               // rare boundary wave
    }
}

// Fast path: trig-free main loop reading the packed precomputed table.
__global__ __launch_bounds__(BLOCK)
void dihedral_energy_tab(const float* __restrict__ pos,
                         const int*   __restrict__ atom_types,
                         const int*   __restrict__ mapping,
                         const int*   __restrict__ mapping_batch,
                         const float* __restrict__ tab,
                         float*       __restrict__ out)
{
    __shared__ float bins[N_BATCH];
    for (int i = threadIdx.x; i < N_BATCH; i += BLOCK) bins[i] = 0.0f;
    __syncthreads();

    const int* __restrict__ m0 = mapping;
    const int* __restrict__ m1 = mapping + (size_t)N_DIHEDRALS;
    const int* __restrict__ m2 = mapping + (size_t)2 * N_DIHEDRALS;
    const int* __restrict__ m3 = mapping + (size_t)3 * N_DIHEDRALS;

    const int base = blockIdx.x * CHUNK;

    #pragma unroll 1
    for (int k = 0; k < ITER; ++k) {
        const int  iraw  = base + k * BLOCK + (int)threadIdx.x;
        const bool valid = iraw < N_DIHEDRALS;
        const int  i     = valid ? iraw : (N_DIHEDRALS - 1);   // clamp: keep EXEC full

        if (iraw + BLOCK < N_DIHEDRALS) {
            __builtin_prefetch(&m0[iraw + BLOCK], 0, 3);       // global_prefetch_b8
            __builtin_prefetch(&m1[iraw + BLOCK], 0, 3);
            __builtin_prefetch(&m2[iraw + BLOCK], 0, 3);
            __builtin_prefetch(&m3[iraw + BLOCK], 0, 3);
            __builtin_prefetch(&mapping_batch[iraw + BLOCK], 0, 3);
        }

        const int a0 = m0[i], a1 = m1[i], a2 = m2[i], a3 = m3[i];

        float c1, s1;
        dihedral_cs(pos, a0, a1, a2, a3, c1, s1);

        // Chebyshev: cos/sin(2t), cos/sin(3t) — pure FMAs.
        const float c2 = fmaf(2.0f * c1, c1, -1.0f);
        const float s2 = 2.0f * s1 * c1;
        const float c3 = c1 * c2 - s1 * s2;
        const float s3 = s1 * c2 + c1 * s2;

        const int t0 = atom_types[a0];
        const int t1 = atom_types[a1];
        const int t2 = atom_types[a2];
        const int t3 = atom_types[a3];
        const int idx = ((t0 * N_TYPES + t1) * N_TYPES + t2) * N_TYPES + t3;

        // Two b128 gathers (entry is 32-B aligned).
        const float4* tp = (const float4*)(tab + (size_t)idx * TAB_STRIDE);
        const float4 lo = tp[0];
        const float4 hi = tp[1];

        // V = ksum - sum_d (c_d * k_d cos th_d + s_d * k_d sin th_d)
        float acc;
        acc = c1 * lo.y + s1 * lo.z;
        acc = fmaf(c2, lo.w, acc);
        acc = fmaf(s2, hi.x, acc);
        acc = fmaf(c3, hi.y, acc);
        acc = fmaf(s3, hi.z, acc);
        float V = lo.x - acc;
        V = valid ? V : 0.0f;

        wave_bin_add(bins, mapping_batch[i], V);
    }

    __syncthreads();
    for (int i = threadIdx.x; i < N_BATCH; i += BLOCK) {
        const float v = bins[i];
        if (v != 0.0f) unsafeAtomicAdd(&out[i], v);  // global_atomic_add_f32
    }
}

// Fallback path (workspace too small): fused kernel with fast hw trig.
__global__ __launch_bounds__(BLOCK)
void dihedral_energy_fused(const float* __restrict__ pos,
                           const int*   __restrict__ atom_types,
                           const int*   __restrict__ mapping,
                           const int*   __restrict__ mapping_batch,
                           const float* __restrict__ thetas,
                           const float* __restrict__ ks,
                           float*       __restrict__ out)
{
    __shared__ float bins[N_BATCH];
    for (int i = threadIdx.x; i < N_BATCH; i += BLOCK) bins[i] = 0.0f;
    __syncthreads();

    const int* __restrict__ m0 = mapping;
    const int* __restrict__ m1 = mapping + (size_t)N_DIHEDRALS;
    const int* __restrict__ m2 = mapping + (size_t)2 * N_DIHEDRALS;
    const int* __restrict__ m3 = mapping + (size_t)3 * N_DIHEDRALS;

    const int base = blockIdx.x * CHUNK;

    #pragma unroll 1
    for (int k = 0; k < ITER; ++k) {
        const int  iraw  = base + k * BLOCK + (int)threadIdx.x;
        const bool valid = iraw < N_DIHEDRALS;
        const int  i     = valid ? iraw : (N_DIHEDRALS - 1);

        const int a0 = m0[i], a1 = m1[i], a2 = m2[i], a3 = m3[i];

        float c1, s1;
        dihedral_cs(pos, a0, a1, a2, a3, c1, s1);

        const float c2 = fmaf(2.0f * c1, c1, -1.0f);
        const float s2 = 2.0f * s1 * c1;
        const float c3 = c1 * c2 - s1 * s2;
        const float s3 = s1 * c2 + c1 * s2;

        const int t0 = atom_types[a0];
        const int t1 = atom_types[a1];
        const int t2 = atom_types[a2];
        const int t3 = atom_types[a3];
        const int idx = ((t0 * N_TYPES + t1) * N_TYPES + t2) * N_TYPES + t3;

        const float th0 = thetas[idx];
        const float th1 = thetas[TBL + idx];
        const float th2 = thetas[2 * TBL + idx];
        const float k0  = ks[idx];
        const float k1  = ks[TBL + idx];
        const float k2  = ks[2 * TBL + idx];

        float sa, ca, sb, cb, sc, cc;
        __sincosf(th0, &sa, &ca);   // hw v_sin/v_cos, |th| <= pi
        __sincosf(th1, &sb, &cb);
        __sincosf(th2, &sc, &cc);

        float V;
        V  = k0 * (1.0f - (c1 * ca + s1 * sa));
        V += k1 * (1.0f - (c2 * cb + s2 * sb));
        V += k2 * (1.0f - (c3 * cc + s3 * sc));
        V = valid ? V : 0.0f;

        wave_bin_add(bins, mapping_batch[i], V);
    }

    __syncthreads();
    for (int i = threadIdx.x; i < N_BATCH; i += BLOCK) {
        const float v = bins[i];
        if (v != 0.0f) unsafeAtomicAdd(&out[i], v);
    }
}

extern "C" void kernel_launch(void* const* d_in, const int* in_sizes, int n_in,
                              void* d_out, int out_size, void* d_ws, size_t ws_size,
                              hipStream_t stream) {
    (void)in_sizes; (void)n_in; (void)out_size;

    const float* pos           = (const float*)d_in[0];
    const int*   atom_types    = (const int*)  d_in[1];
    const int*   mapping       = (const int*)  d_in[2];
    const int*   mapping_batch = (const int*)  d_in[3];
    const float* thetas        = (const float*)d_in[4];
    const float* ks            = (const float*)d_in[5];
    float*       out           = (float*)d_out;

    dihedral_zero_out<<<1, N_BATCH, 0, stream>>>(out);

    const int grid = (N_DIHEDRALS + CHUNK - 1) / CHUNK;  // 1954 blocks

    if (ws_size >= TAB_BYTES) {
        float* tab = (float*)d_ws;
        dihedral_build_table<<<(TBL + 255) / 256, 256, 0, stream>>>(thetas, ks, tab);
        dihedral_energy_tab<<<grid, BLOCK, 0, stream>>>(
            pos, atom_types, mapping, mapping_batch, tab, out);
    } else {
        dihedral_energy_fused<<<grid, BLOCK, 0, stream>>>(
            pos, atom_types, mapping, mapping_batch, thetas, ks, out);
    }
}